// MultiHeadSelfAttention_83519934038321
// MI455X (gfx1250) — compile-verified
//
#include <hip/hip_runtime.h>
#include <hip/hip_bf16.h>

typedef __attribute__((ext_vector_type(2))) float v2f;
typedef __attribute__((ext_vector_type(8))) float v8f;

#define D_MODEL 1024
#define NHEAD 16
#define DK 64

// D = A(16x4) * B(4x16) + C, full fp32 (V_WMMA_F32_16X16X4_F32)
__device__ __forceinline__ v8f wmma_f32(v2f a, v2f b, v8f c) {
  return __builtin_amdgcn_wmma_f32_16x16x4_f32(
      /*neg_a=*/false, a, /*neg_b=*/false, b,
      /*c_mod=*/(short)0, c, /*reuse_a=*/false, /*reuse_b=*/false);
}

// CDNA5 async global->LDS copy: 16B per lane, no VGPR staging, ASYNCcnt.
__device__ __forceinline__ void async_copy_b128(const float* gsrc, const float* lds_dst) {
  unsigned long long ga = (unsigned long long)(uintptr_t)gsrc;
  unsigned la = (unsigned)(uintptr_t)lds_dst;   // flat addr low 32 = LDS address
  asm volatile("global_load_async_to_lds_b128 %0, %1, off"
               :: "v"(la), "v"(ga) : "memory");
}
__device__ __forceinline__ void async_wait0() {
  asm volatile("s_wait_asynccnt 0x0" ::: "memory");
}

// ---------------------------------------------------------------------------
// Tiled GEMM: out = A[M x 1024] @ W[1024 x 1024] + bias
// Block tile 128x64, K-step 32, double-buffered LDS (1 barrier / K-tile).
// A tile: GLOBAL_LOAD_ASYNC_TO_LDS_B128 (no staging VGPRs -> no spills).
// W tile: staged through 8 VGPRs and stored TRANSPOSED so B-fragments are
// contiguous b64 loads.
// mode 0: out[row*1024 + col]
// mode 1: out stored in [B, H, T, dk] layout (row=(b,t), col=(h,d))
// ---------------------------------------------------------------------------
#define GM_BM 128
#define GM_BN 64
#define GM_BK 32
#define AS_LD 36   // row stride: 4*(9m mod 16) -> 16 distinct banks, 16B-aligned
#define WT_LD 36   // transposed W tile Wt[n][k], 64 rows x 32 cols

__global__ __launch_bounds__(256) __attribute__((amdgpu_waves_per_eu(1, 4)))
void proj_gemm_kernel(
    const float* __restrict__ A, const float* __restrict__ W,
    const float* __restrict__ bias, float* __restrict__ out,
    int M, int mode, int T)
{
  __shared__ __align__(16) float As[2][GM_BM * AS_LD];
  __shared__ __align__(16) float Wt[2][GM_BN * WT_LD];

  const int tid  = threadIdx.x;
  const int w    = tid >> 5;
  const int lane = tid & 31;
  const int half = lane >> 4;
  const int l16  = lane & 15;
  const int wm   = w & 3;        // 4 row-bands of 32
  const int wn   = w >> 2;       // 2 col-bands of 32
  const int rowBase = blockIdx.x * GM_BM;
  const int colBase = blockIdx.y * GM_BN;

  float4 rw[2];                  // W staging (transpose on LDS store)

  auto gloadA = [&](int k0, int buf) {     // async, no VGPR staging
    #pragma unroll
    for (int p = 0; p < 4; ++p) {
      int flat = tid * 4 + p * 1024;
      int r = flat >> 5, c = flat & 31;
      async_copy_b128(A + (size_t)(rowBase + r) * D_MODEL + k0 + c,
                      &As[buf][r * AS_LD + c]);
    }
  };
  auto gloadW = [&](int k0) {
    #pragma unroll
    for (int p = 0; p < 2; ++p) {
      int flat = tid * 4 + p * 1024;
      int r = flat >> 6, c = flat & 63;
      rw[p] = *(const float4*)(W + (size_t)(k0 + r) * D_MODEL + colBase + c);
    }
  };
  auto sstoreW = [&](int buf) {
    #pragma unroll
    for (int p = 0; p < 2; ++p) {
      int flat = tid * 4 + p * 1024;
      int r = flat >> 6, c = flat & 63;
      Wt[buf][(c + 0) * WT_LD + r] = rw[p].x;   // transpose into LDS
      Wt[buf][(c + 1) * WT_LD + r] = rw[p].y;
      Wt[buf][(c + 2) * WT_LD + r] = rw[p].z;
      Wt[buf][(c + 3) * WT_LD + r] = rw[p].w;
    }
  };

  v8f acc[2][2];
  #pragma unroll
  for (int i = 0; i < 2; ++i)
    #pragma unroll
    for (int j = 0; j < 2; ++j) acc[i][j] = {};

  gloadA(0, 0);
  gloadW(0);
  sstoreW(0);

  const int NT = D_MODEL / GM_BK;   // 32 K-tiles
  for (int t = 0; t < NT; ++t) {
    async_wait0();                  // this wave's async LDS writes done
    __syncthreads();                // all waves' tile writes published
    const int cur = t & 1;
    if (t + 1 < NT) {
      gloadA((t + 1) * GM_BK, 1 - cur);   // async, runs during compute
      gloadW((t + 1) * GM_BK);            // in flight during compute
    }

    const float* as = &As[cur][0];
    const float* wt = &Wt[cur][0];

    // software-pipelined fragment loads (prefetch kk+1 before WMMAs of kk)
    v2f afr[2][2], bfr[2][2];
    {
      const int kb = 2 * half;
      #pragma unroll
      for (int mi = 0; mi < 2; ++mi)
        afr[0][mi] = *(const v2f*)(as + (wm * 32 + mi * 16 + l16) * AS_LD + kb);
      #pragma unroll
      for (int ni = 0; ni < 2; ++ni)
        bfr[0][ni] = *(const v2f*)(wt + (wn * 32 + ni * 16 + l16) * WT_LD + kb);
    }
    #pragma unroll
    for (int kk = 0; kk < GM_BK / 4; ++kk) {
      const int c = kk & 1, nx = c ^ 1;
      if (kk + 1 < GM_BK / 4) {
        const int kb = (kk + 1) * 4 + 2 * half;
        #pragma unroll
        for (int mi = 0; mi < 2; ++mi)
          afr[nx][mi] = *(const v2f*)(as + (wm * 32 + mi * 16 + l16) * AS_LD + kb);
        #pragma unroll
        for (int ni = 0; ni < 2; ++ni)
          bfr[nx][ni] = *(const v2f*)(wt + (wn * 32 + ni * 16 + l16) * WT_LD + kb);
      }
      #pragma unroll
      for (int mi = 0; mi < 2; ++mi)
        #pragma unroll
        for (int ni = 0; ni < 2; ++ni)
          acc[mi][ni] = wmma_f32(afr[c][mi], bfr[c][ni], acc[mi][ni]);
    }

    if (t + 1 < NT) sstoreW(1 - cur);
  }

  // store (C/D layout: VGPR r -> row r + 8*half, col = l%16)
  #pragma unroll
  for (int mi = 0; mi < 2; ++mi) {
    #pragma unroll
    for (int ni = 0; ni < 2; ++ni) {
      int colw = colBase + wn * 32 + ni * 16 + l16;
      float bval = bias[colw];
      #pragma unroll
      for (int r = 0; r < 8; ++r) {
        int row = rowBase + wm * 32 + mi * 16 + r + 8 * half;
        float val = acc[mi][ni][r] + bval;
        if (mode == 0) {
          out[(size_t)row * D_MODEL + colw] = val;
        } else {
          int b_ = row / T, t_ = row % T;
          int h_ = colw >> 6, d_ = colw & 63;
          out[(((size_t)b_ * NHEAD + h_) * T + t_) * DK + d_] = val;
        }
      }
    }
  }
}

// ---------------------------------------------------------------------------
// Flash attention: grid (T/128, B*H). 8 waves; wave w owns query rows
// qBase + 16w .. +15. Streams K/V in double-buffered tiles of 64 keys with
// online softmax. K tile loaded with GLOBAL_LOAD_ASYNC_TO_LDS_B128 (no
// staging regs); V tile staged in 16 VGPRs and stored TRANSPOSED (Vt[d][key])
// for contiguous B-fragments. q/k/v in [B,H,T,dk]; ctx in [B,T,H,dk].
// ---------------------------------------------------------------------------
#define KV_LD 68   // 64 + 4 pad: lane-indexed rows -> distinct banks, 8B-aligned

__global__ __launch_bounds__(256) __attribute__((amdgpu_waves_per_eu(1, 4)))
void flash_attn_kernel(
    const float* __restrict__ q, const float* __restrict__ k,
    const float* __restrict__ v, const int* __restrict__ mask,
    float* __restrict__ ctx, int T)
{
  __shared__ __align__(16) float Ks[2][64 * KV_LD];   // row-major [key][d]
  __shared__ __align__(16) float Vt[2][64 * KV_LD];   // transposed [d][key]
  __shared__ __align__(16) float Ps[8 * 16 * KV_LD];  // per-wave P staging
  __shared__ int Ms[2][64];

  const int tid  = threadIdx.x;
  const int w    = tid >> 5;
  const int lane = tid & 31;
  const int half = lane >> 4;
  const int l16  = lane & 15;
  const int bh   = blockIdx.y;
  const int b    = bh / NHEAD;
  const int h    = bh % NHEAD;
  const int qBase = blockIdx.x * 128;
  const float scale = 0.125f;                 // 1/sqrt(dk)

  const float* kb_g = k + (size_t)bh * T * DK;
  const float* vb_g = v + (size_t)bh * T * DK;
  const int pOff = w * 16 * KV_LD;

  float4 rv[4];
  int rm = 1;

  auto gloadK = [&](int t0, int buf) {      // async, no VGPR staging
    #pragma unroll
    for (int p = 0; p < 4; ++p) {
      int flat = tid * 4 + p * 1024;
      int r = flat >> 6, c = flat & 63;
      async_copy_b128(kb_g + (size_t)(t0 + r) * DK + c, &Ks[buf][r * KV_LD + c]);
    }
  };
  auto gloadV = [&](int t0) {
    #pragma unroll
    for (int p = 0; p < 4; ++p) {
      int flat = tid * 4 + p * 1024;
      int r = flat >> 6, c = flat & 63;
      rv[p] = *(const float4*)(vb_g + (size_t)(t0 + r) * DK + c);
    }
    if (tid < 64) rm = mask[(size_t)b * T + t0 + tid];
  };
  auto sstoreV = [&](int buf) {
    #pragma unroll
    for (int p = 0; p < 4; ++p) {
      int flat = tid * 4 + p * 1024;
      int r = flat >> 6, c = flat & 63;
      Vt[buf][(c + 0) * KV_LD + r] = rv[p].x;   // transpose into LDS
      Vt[buf][(c + 1) * KV_LD + r] = rv[p].y;
      Vt[buf][(c + 2) * KV_LD + r] = rv[p].z;
      Vt[buf][(c + 3) * KV_LD + r] = rv[p].w;
    }
    if (tid < 64) Ms[buf][tid] = rm;
  };

  // Q fragments for 16 chained 16x16x4 WMMAs over dk=64, pre-scaled
  v2f aQ[16];
  {
    const float* qp = q + ((size_t)bh * T + qBase + w * 16 + l16) * DK;
    #pragma unroll
    for (int kk = 0; kk < 16; ++kk) {
      int kb = kk * 4 + 2 * half;
      aQ[kk].x = qp[kb] * scale;
      aQ[kk].y = qp[kb + 1] * scale;
    }
  }

  float mrow[8], lrow[8];
  v8f o[4];
  #pragma unroll
  for (int r = 0; r < 8; ++r) { mrow[r] = -3.0e38f; lrow[r] = 0.0f; }
  #pragma unroll
  for (int jo = 0; jo < 4; ++jo) o[jo] = {};

  gloadK(0, 0);
  gloadV(0);
  sstoreV(0);

  const int NT = T / 64;
  for (int it = 0; it < NT; ++it) {
    async_wait0();                  // this wave's async K writes done
    __syncthreads();                // tile published to all waves
    const int cur = it & 1;
    if (it + 1 < NT) {
      gloadK((it + 1) * 64, 1 - cur);   // async, runs during compute
      gloadV((it + 1) * 64);            // in flight during scores/softmax
    }

    const float* ks = &Ks[cur][0];
    const float* vt = &Vt[cur][0];

    // S[16 x 64] = (Q*scale) @ K^T : 4 col-tiles x 16 chained WMMAs, pipelined
    v8f s[4];
    #pragma unroll
    for (int j = 0; j < 4; ++j) {
      const int key = j * 16 + l16;
      v8f acc = {};
      v2f bk[2];
      bk[0] = *(const v2f*)(ks + key * KV_LD + 2 * half);
      #pragma unroll
      for (int kk = 0; kk < 16; ++kk) {
        const int c = kk & 1, nx = c ^ 1;
        if (kk + 1 < 16)
          bk[nx] = *(const v2f*)(ks + key * KV_LD + (kk + 1) * 4 + 2 * half);
        acc = wmma_f32(aQ[kk], bk[c], acc);
      }
      s[j] = acc;
    }

    // mask (broadcast over queries): score -> -1e9 where mask==0
    #pragma unroll
    for (int j = 0; j < 4; ++j) {
      if (Ms[cur][j * 16 + l16] == 0) {
        #pragma unroll
        for (int r = 0; r < 8; ++r) s[j][r] = -1.0e9f;
      }
    }

    // online softmax per row (row m = r + 8*half, cols spread over 16 lanes)
    #pragma unroll
    for (int r = 0; r < 8; ++r) {
      float t = fmaxf(fmaxf(s[0][r], s[1][r]), fmaxf(s[2][r], s[3][r]));
      t = fmaxf(t, __shfl_xor(t, 1, 32));
      t = fmaxf(t, __shfl_xor(t, 2, 32));
      t = fmaxf(t, __shfl_xor(t, 4, 32));
      t = fmaxf(t, __shfl_xor(t, 8, 32));
      float mnew = fmaxf(mrow[r], t);
      float sc = __expf(mrow[r] - mnew);
      lrow[r] *= sc;
      #pragma unroll
      for (int jo = 0; jo < 4; ++jo) o[jo][r] *= sc;
      float rsum = 0.0f;
      #pragma unroll
      for (int j = 0; j < 4; ++j) {
        float p = __expf(s[j][r] - mnew);
        rsum += p;
        // C-layout -> LDS (row r+8*half, col j*16+l16) for A-layout reload
        Ps[pOff + (r + 8 * half) * KV_LD + j * 16 + l16] = p;
      }
      rsum += __shfl_xor(rsum, 1, 32);
      rsum += __shfl_xor(rsum, 2, 32);
      rsum += __shfl_xor(rsum, 4, 32);
      rsum += __shfl_xor(rsum, 8, 32);
      lrow[r] += rsum;
      mrow[r] = mnew;
    }

    // park next V tile into the other buffer (frees staging regs)
    if (it + 1 < NT) sstoreV(1 - cur);

    // O += P @ V  (Ps region is per-wave: wave-ordered LDS, no barrier needed)
    #pragma unroll
    for (int jo = 0; jo < 4; ++jo) {
      const int drow = jo * 16 + l16;
      v2f ap[2], bv[2];
      ap[0] = *(const v2f*)(&Ps[pOff + l16 * KV_LD + 2 * half]);
      bv[0] = *(const v2f*)(vt + drow * KV_LD + 2 * half);
      #pragma unroll
      for (int kk = 0; kk < 16; ++kk) {
        const int c = kk & 1, nx = c ^ 1;
        if (kk + 1 < 16) {
          const int kb = (kk + 1) * 4 + 2 * half;
          ap[nx] = *(const v2f*)(&Ps[pOff + l16 * KV_LD + kb]);
          bv[nx] = *(const v2f*)(vt + drow * KV_LD + kb);
        }
        o[jo] = wmma_f32(ap[c], bv[c], o[jo]);
      }
    }
  }

  // epilogue: normalize and store ctx in [B, T, H, dk]
  #pragma unroll
  for (int r = 0; r < 8; ++r) {
    float inv = 1.0f / lrow[r];
    int t = qBase + w * 16 + r + 8 * half;
    #pragma unroll
    for (int jo = 0; jo < 4; ++jo) {
      int d = jo * 16 + l16;
      ctx[(((size_t)b * T + t) * NHEAD + h) * DK + d] = o[jo][r] * inv;
    }
  }
}

// ---------------------------------------------------------------------------
extern "C" void kernel_launch(void* const* d_in, const int* in_sizes, int n_in,
                              void* d_out, int out_size, void* d_ws, size_t ws_size,
                              hipStream_t stream) {
  const float* x    = (const float*)d_in[0];
  const int*   mask = (const int*)  d_in[1];
  const float* Wq   = (const float*)d_in[2];
  const float* bq   = (const float*)d_in[3];
  const float* Wk   = (const float*)d_in[4];
  const float* bk   = (const float*)d_in[5];
  const float* Wv   = (const float*)d_in[6];
  const float* bv   = (const float*)d_in[7];
  const float* Wo   = (const float*)d_in[8];
  const float* bo   = (const float*)d_in[9];
  float* out = (float*)d_out;

  const int BT = in_sizes[0] / D_MODEL;   // B*T = 8192
  const int T  = 2048;
  const int B  = BT / T;

  float* qb  = (float*)d_ws;
  float* kb  = qb + (size_t)BT * D_MODEL;
  float* vb  = kb + (size_t)BT * D_MODEL;
  float* cxb = vb + (size_t)BT * D_MODEL;

  dim3 blk(256);
  dim3 gemmGrid(BT / GM_BM, D_MODEL / GM_BN);   // 64 x 16

  // Q/K/V projections -> [B,H,T,dk]
  proj_gemm_kernel<<<gemmGrid, blk, 0, stream>>>(x, Wq, bq, qb, BT, 1, T);
  proj_gemm_kernel<<<gemmGrid, blk, 0, stream>>>(x, Wk, bk, kb, BT, 1, T);
  proj_gemm_kernel<<<gemmGrid, blk, 0, stream>>>(x, Wv, bv, vb, BT, 1, T);

  // attention -> ctx [B,T,H,dk]
  dim3 attnGrid(T / 128, B * NHEAD);            // 16 x 64
  flash_attn_kernel<<<attnGrid, blk, 0, stream>>>(qb, kb, vb, mask, cxb, T);

  // output projection -> [B,T,D]
  proj_gemm_kernel<<<gemmGrid, blk, 0, stream>>>(cxb, Wo, bo, out, BT, 0, T);
}